// KnotForward_35923106464289
// MI455X (gfx1250) — compile-verified
//
#include <hip/hip_runtime.h>
#include <hip/hip_bf16.h>

// One workgroup per batch element. 128 threads = 4 wave32 waves.
// All state (knots, z, GEMM operands, insertion scratch) lives in LDS (~53KB).
// Matrix layers use V_WMMA_F32_16X16X4_F32 (f32 needed: eps=1e-6 crossing tests).

#define BATCH    512
#define HDIM     64
#define DDIM     3
#define MAXK     64
#define KPAD     68            // 64 relu(z) + 3 coords + 1 zero pad, multiple of 4
#define EPSV     1e-6f
#define NTHREADS 128

typedef __attribute__((ext_vector_type(2))) float v2f;
typedef __attribute__((ext_vector_type(8))) float v8f;

// ---------------------------------------------------------------------------
// Zero-crossing insertion for one block. Validity is a prefix of length *s_n.
// bufA: 64*KPAD floats scratch (candidate t here; GEMM A-matrix elsewhere)
// bufB: KPAD*64 floats scratch (ranks, then recomputed z; GEMM B elsewhere)
// ---------------------------------------------------------------------------
__device__ __forceinline__ void insert_zero_crossings_blk(
    float* s_t, int* s_n, float* s_z,
    float* bufA, float* bufB,
    int* s_m, int* s_off, int* s_ntot,
    float* out_t, int* out_seg, float* out_al, int tid)
{
  float* cand_t = bufA;          // [63][64] candidate crossing times (2.0 = invalid)
  int*   rank   = (int*)bufB;    // [63][64] stable rank among segment candidates

  if (tid < MAXK) s_m[tid] = 0;
  __syncthreads();
  const int n = *s_n;

  // 1) candidates + per-segment valid counts (LDS atomics -> ds_add)
  for (int i = tid; i < 63 * 64; i += NTHREADS) {
    const int s = i >> 6, c = i & 63;
    float tc = 2.0f;
    if (s < n - 1) {
      const float zL = s_z[s * 64 + c];
      const float zR = s_z[(s + 1) * 64 + c];
      const float den = zR - zL;
      if (fabsf(den) > EPSV) {
        const float al = -zL / den;
        if (al > EPSV && al < 1.0f - EPSV) {
          tc = fmaf(al, s_t[s + 1] - s_t[s], s_t[s]);
          atomicAdd(&s_m[s], 1);
        }
      }
    }
    cand_t[i] = tc;
  }
  __syncthreads();

  // 2) stable ranking sort within each segment (valid t < 1 < 2.0 invalid)
  for (int i = tid; i < 63 * 64; i += NTHREADS) {
    const int s = i >> 6, c = i & 63;
    int r = -1;
    if (s < n - 1 && cand_t[i] < 1.5f) {
      const float ti = cand_t[i];
      const float* seg = &cand_t[s * 64];
      r = 0;
      for (int j = 0; j < 64; ++j) {
        const float tj = seg[j];
        r += (tj < ti) || (tj == ti && j < c);
      }
    }
    rank[i] = r;
  }
  __syncthreads();

  // 3) per-segment stream offsets (tiny serial scan) + clear outputs
  if (tid == 0) {
    int run = 0;
    for (int s = 0; s < 64; ++s) {
      s_off[s] = run;
      run += ((s < n) ? 1 : 0) + s_m[s];
    }
    *s_ntot = run < MAXK ? run : MAXK;
  }
  if (tid < MAXK) { out_t[tid] = 1.0f; out_seg[tid] = 0; out_al[tid] = 0.0f; }
  __syncthreads();

  // 4) scatter surviving entries (provenance: segment + alpha)
  if (tid < 64 && tid < n) {                       // original knots
    const int idx = s_off[tid];
    if (idx < MAXK) { out_t[idx] = s_t[tid]; out_seg[idx] = tid; out_al[idx] = 0.0f; }
  }
  for (int i = tid; i < 63 * 64; i += NTHREADS) {  // crossings, in sorted order
    const int r = rank[i];
    if (r >= 0) {
      const int s = i >> 6, c = i & 63;
      const int idx = s_off[s] + 1 + r;
      if (idx < MAXK) {
        const float zL = s_z[s * 64 + c];
        const float den = s_z[(s + 1) * 64 + c] - zL;
        out_t[idx]  = cand_t[i];
        out_seg[idx] = s;
        out_al[idx]  = -zL / den;
      }
    }
  }
  __syncthreads();

  // 5) recompute z rows at surviving knots (bufB reused: ranks are dead)
  float* znew = bufB;
  const int nn = *s_ntot;
  for (int i = tid; i < 64 * 64; i += NTHREADS) {
    const int j = i >> 6, c = i & 63;
    float v = 0.0f;
    if (j < nn) {
      const int s  = out_seg[j];
      const int s1 = (s + 1 < 64) ? s + 1 : 63;   // alpha==0 for copies; clamp OK
      const float zL = s_z[s * 64 + c];
      v = fmaf(out_al[j], s_z[s1 * 64 + c] - zL, zL);
    }
    znew[i] = v;
  }
  __syncthreads();
  for (int i = tid; i < 64 * 64; i += NTHREADS) s_z[i] = znew[i];
  if (tid < MAXK) s_t[tid] = out_t[tid];          // invalid positions stay 1.0
  if (tid == 0) *s_n = nn;
  __syncthreads();
}

// ---------------------------------------------------------------------------
// z = [relu(z), coords] @ W^T + b  as 64x68 @ 68x64 via V_WMMA_F32_16X16X4_F32.
// wave w owns tile-row w (16 rows), 4 column tiles, 17 K=4 steps.
// ---------------------------------------------------------------------------
__device__ __forceinline__ void affine_layer_wmma(
    const float* __restrict__ Wg, const float* __restrict__ bg,
    float* hin, float* wt, float* s_z, const float* s_t,
    const float* s_start, const float* s_d, const int* s_n, int tid)
{
  const int n = *s_n;
  // A: h_input 64 x KPAD row-major
  for (int i = tid; i < 64 * KPAD; i += NTHREADS) {
    const int j = i / KPAD, k = i - j * KPAD;
    float v = 0.0f;
    if (j < n) {
      if (k < 64)      { const float z = s_z[j * 64 + k]; v = z > 0.0f ? z : 0.0f; }
      else if (k < 67) { const int dd = k - 64; v = fmaf(s_t[j], s_d[dd], s_start[dd]); }
    }
    hin[i] = v;
  }
  // B: W^T  KPAD x 64 (k-major), zero-padded K row
  for (int i = tid; i < KPAD * 64; i += NTHREADS) {
    const int k = i >> 6, h = i & 63;
    wt[i] = (k < 67) ? Wg[h * 67 + k] : 0.0f;
  }
  __syncthreads();

  const int mi   = tid >> 5;            // wave id = tile row
  const int lane = tid & 31;
  const int lrow = mi * 16 + (lane & 15);
  const int kh   = (lane >> 4) * 2;     // A/B f32 K=4 layout: half-wave K split
  v8f acc[4] = {};
  for (int k0 = 0; k0 < KPAD; k0 += 4) {
    v2f a;
    a.x = hin[lrow * KPAD + k0 + kh];
    a.y = hin[lrow * KPAD + k0 + kh + 1];
#pragma unroll
    for (int ni = 0; ni < 4; ++ni) {
      const int col = ni * 16 + (lane & 15);
      v2f bb;
      bb.x = wt[(k0 + kh) * 64 + col];
      bb.y = wt[(k0 + kh + 1) * 64 + col];
#if defined(__gfx1250__) && __has_builtin(__builtin_amdgcn_wmma_f32_16x16x4_f32)
      acc[ni] = __builtin_amdgcn_wmma_f32_16x16x4_f32(
          false, a, false, bb, (short)0, acc[ni], false, false);
#else
      acc[ni][0] += a.x * bb.x + a.y * bb.y;   // host-pass placeholder only
#endif
    }
  }
  __syncthreads();

  // C/D layout: VGPR r -> M = r (lanes 0-15) / r+8 (lanes 16-31), N = lane&15
  const int half = (lane >> 4) * 8;
#pragma unroll
  for (int ni = 0; ni < 4; ++ni) {
    const int col = ni * 16 + (lane & 15);
    const float bias = bg[col];
#pragma unroll
    for (int r = 0; r < 8; ++r) {
      const int mrow = mi * 16 + half + r;
      s_z[mrow * 64 + col] = (mrow < n) ? acc[ni][r] + bias : 0.0f;
    }
  }
  __syncthreads();
}

// ---------------------------------------------------------------------------
__global__ __launch_bounds__(NTHREADS) void knot_forward_kernel(
    const float* __restrict__ endp, const float* __restrict__ startp,
    const float* __restrict__ W0, const float* __restrict__ b0,
    const float* __restrict__ W1, const float* __restrict__ b1,
    const float* __restrict__ W2, const float* __restrict__ b2,
    const float* __restrict__ W3, const float* __restrict__ b3,
    float* __restrict__ out)
{
  __shared__ float s_z[64 * 64];          // 16 KB  knot x channel values
  __shared__ float s_bufA[64 * KPAD];     // 17 KB  cand_t / GEMM A
  __shared__ float s_bufB[KPAD * 64];     // 17 KB  ranks+znew / GEMM B
  __shared__ float s_t[64];
  __shared__ float s_outT[64];
  __shared__ int   s_outSeg[64];
  __shared__ float s_outAl[64];
  __shared__ int   s_m[64], s_off[64];
  __shared__ int   s_n, s_ntot;
  __shared__ float s_start[DDIM], s_d[DDIM];
  __shared__ float s_a0[64], s_b0[64];

  const int b = blockIdx.x, tid = threadIdx.x;
  if (tid == 0) {                         // global_prefetch_b8 for the weights
    __builtin_prefetch(W1, 0, 3);
    __builtin_prefetch(W2, 0, 3);
    __builtin_prefetch(W3, 0, 3);
  }
  if (tid < DDIM) {
    const float sp = startp[b * DDIM + tid];
    s_start[tid] = sp;
    s_d[tid] = endp[b * DDIM + tid] - sp;
  }
  __syncthreads();

  if (tid < 64) {
    float a = 0.0f, be = b0[tid];
    for (int dd = 0; dd < DDIM; ++dd) {
      a  = fmaf(W0[tid * DDIM + dd], s_d[dd], a);
      be = fmaf(W0[tid * DDIM + dd], s_start[dd], be);
    }
    s_a0[tid] = a; s_b0[tid] = be;
    s_t[tid] = (tid == 0) ? 0.0f : 1.0f;  // t = [0, 1, 1, 1, ...]
  }
  if (tid == 0) s_n = 2;
  __syncthreads();

  // layer-0 z is affine in t (rank-1): no GEMM needed
  for (int i = tid; i < 64 * 64; i += NTHREADS) {
    const int j = i >> 6, c = i & 63;
    s_z[i] = (j < 2) ? fmaf(s_a0[c], s_t[j], s_b0[c]) : 0.0f;
  }
  __syncthreads();

  insert_zero_crossings_blk(s_t, &s_n, s_z, s_bufA, s_bufB, s_m, s_off, &s_ntot,
                            s_outT, s_outSeg, s_outAl, tid);
  affine_layer_wmma(W1, b1, s_bufA, s_bufB, s_z, s_t, s_start, s_d, &s_n, tid);
  insert_zero_crossings_blk(s_t, &s_n, s_z, s_bufA, s_bufB, s_m, s_off, &s_ntot,
                            s_outT, s_outSeg, s_outAl, tid);
  affine_layer_wmma(W2, b2, s_bufA, s_bufB, s_z, s_t, s_start, s_d, &s_n, tid);
  insert_zero_crossings_blk(s_t, &s_n, s_z, s_bufA, s_bufB, s_m, s_off, &s_ntot,
                            s_outT, s_outSeg, s_outAl, tid);

  // head: y = relu(z) @ W3^T + b3, masked; outputs (t | valid | y) flat
  const int n = s_n;
  if (tid < 64) {
    out[(size_t)b * 64 + tid] = s_t[tid];
    out[(size_t)(BATCH * 64) + (size_t)b * 64 + tid] = (tid < n) ? 1.0f : 0.0f;
    float acc = 0.0f;
    for (int c = 0; c < 64; ++c) {
      const float z = s_z[tid * 64 + c];
      acc = fmaf(z > 0.0f ? z : 0.0f, W3[c], acc);
    }
    out[(size_t)(2 * BATCH * 64) + (size_t)b * 64 + tid] =
        (tid < n) ? acc + b3[0] : 0.0f;
  }
}

extern "C" void kernel_launch(void* const* d_in, const int* in_sizes, int n_in,
                              void* d_out, int out_size, void* d_ws, size_t ws_size,
                              hipStream_t stream) {
  (void)in_sizes; (void)n_in; (void)out_size; (void)d_ws; (void)ws_size;
  const float* endp   = (const float*)d_in[0];
  const float* startp = (const float*)d_in[1];
  const float* W0 = (const float*)d_in[2];
  const float* b0 = (const float*)d_in[3];
  const float* W1 = (const float*)d_in[4];
  const float* b1 = (const float*)d_in[5];
  const float* W2 = (const float*)d_in[6];
  const float* b2 = (const float*)d_in[7];
  const float* W3 = (const float*)d_in[8];
  const float* b3 = (const float*)d_in[9];
  float* out = (float*)d_out;
  knot_forward_kernel<<<BATCH, NTHREADS, 0, stream>>>(
      endp, startp, W0, b0, W1, b1, W2, b2, W3, b3, out);
}